// NLayer_39840116637818
// MI455X (gfx1250) — compile-verified
//
#include <hip/hip_runtime.h>
#include <hip/hip_bf16.h>

typedef float v2f __attribute__((ext_vector_type(2)));
typedef float v8f __attribute__((ext_vector_type(8)));

#define BB   2
#define VV   20000
#define CC   32
#define NB   12
#define KS   9
#define OUT  32
#define NAUG 304               // 288 (wx cols) + 9 (ux cols) + 7 pad -> 19 tiles of 16
#define MROWS (BB*VV)          // 40000
#define MT   (MROWS/16)        // 2500 exactly
#define NT   (NAUG/16)         // 19

// ---------------------------------------------------------------------------
// Kernel 0: build augmented weight matrix Waug[32][304]:
//   cols 0..287  = W[c][k][o]  (already k*32+o contiguous in W)
//   cols 288..296= u[c][k]
//   cols 297..303= 0
// ---------------------------------------------------------------------------
__global__ void nl_prep_waug(const float* __restrict__ W,
                             const float* __restrict__ u,
                             float* __restrict__ Waug) {
    int idx = blockIdx.x * blockDim.x + threadIdx.x;
    if (idx >= CC * NAUG) return;
    int c   = idx / NAUG;
    int col = idx - c * NAUG;
    float val;
    if (col < KS * OUT)            val = W[c * (KS * OUT) + col];
    else if (col < KS * OUT + KS)  val = u[c * KS + (col - KS * OUT)];
    else                           val = 0.0f;
    Waug[idx] = val;
}

// ---------------------------------------------------------------------------
// Kernel 1: tall-skinny f32 GEMM via V_WMMA_F32_16X16X4_F32.
//   buf[b*V+v][0..287]   = wx  (k*32+o)
//   buf[b*V+v][288..296] = ux
// One wave per (16-row, 16-col) tile; 8 chained K=4 WMMAs cover K=32.
// ---------------------------------------------------------------------------
__global__ void nl_gemm_wmma(const float* __restrict__ x,     // (B*V, 32)
                             const float* __restrict__ Waug,  // (32, 304)
                             float* __restrict__ buf) {       // (B*V, 304)
    const int wavesPerBlock = blockDim.x >> 5;
    int tile = blockIdx.x * wavesPerBlock + (threadIdx.x >> 5);
    if (tile >= MT * NT) return;               // wave-uniform: EXEC stays all-1s
    const int mt = tile / NT;
    const int nt = tile - mt * NT;
    const int lane = threadIdx.x & 31;
    const int half = lane >> 4;                // 0: K even pair base, 1: +2
    const int l    = lane & 15;

    // A fragments: row mt*16+l of x, K chunks of 4
    const float* xrow = x + (size_t)(mt * 16 + l) * CC;
    v2f a[8];
#pragma unroll
    for (int kk = 0; kk < 8; ++kk) {
        int kb = 4 * kk + 2 * half;
        a[kk] = *(const v2f*)(xrow + kb);      // 8B aligned (kb even, row 128B aligned)
    }

    // B fragments + 8 WMMAs
    const float* wcol = Waug + nt * 16 + l;
    v8f acc = {};
#pragma unroll
    for (int kk = 0; kk < 8; ++kk) {
        int kr = 4 * kk + 2 * half;
        v2f bf;
        bf.x = wcol[(size_t)kr * NAUG];
        bf.y = wcol[(size_t)(kr + 1) * NAUG];
        acc = __builtin_amdgcn_wmma_f32_16x16x4_f32(
            /*neg_a=*/false, a[kk], /*neg_b=*/false, bf,
            /*c_mod=*/(short)0, acc, /*reuse_a=*/false, /*reuse_b=*/false);
    }

    // D: VGPR i -> (M = half*8 + i, N = l)
    float* drow = buf + (size_t)(mt * 16) * NAUG + nt * 16 + l;
#pragma unroll
    for (int i = 0; i < 8; ++i) {
        drow[(size_t)(half * 8 + i) * NAUG] = acc[i];
    }
}

// ---------------------------------------------------------------------------
// Kernel 2: per-(b,v) wave: neighbor gather (L2-resident buf) + softmax(9)
// + weighted aggregation + deg normalization + bias + relu.
// lane = output channel o (OUT==32==wave32).
// ---------------------------------------------------------------------------
__global__ void nl_gather_agg(const float* __restrict__ buf,
                              const int*   __restrict__ adj,
                              const float* __restrict__ cvec,
                              const float* __restrict__ bvec,
                              float*       __restrict__ out) {
    const int wavesPerBlock = blockDim.x >> 5;
    int w = blockIdx.x * wavesPerBlock + (threadIdx.x >> 5);
    if (w >= BB * VV) return;                  // wave-uniform
    const int lane = threadIdx.x & 31;
    const int b = w / VV;
    const int v = w - b * VV;

    const float* self_row = buf + (size_t)w * NAUG;
    float ck[KS], uxk[KS];
#pragma unroll
    for (int k = 0; k < KS; ++k) {
        ck[k]  = cvec[k];
        uxk[k] = self_row[KS * OUT + k];       // ux at cols 288..296
    }
    const float bias = bvec[lane];

    const float* bufb = buf + (size_t)b * VV * NAUG;
    float acc = 0.0f;
    int deg = 0;
#pragma unroll
    for (int n = 0; n < NB; ++n) {
        int j = adj[v * NB + n];               // wave-uniform
        if (j == 0) continue;                  // zero-padded neighbor: contributes 0
        ++deg;
        const float* nrow = bufb + (size_t)(j - 1) * NAUG;

        // softmax over KS of (ux - nb_ux + c), numerically stable
        float t[KS];
        float mx = -3.402823466e38f;
#pragma unroll
        for (int k = 0; k < KS; ++k) {
            t[k] = uxk[k] - nrow[KS * OUT + k] + ck[k];
            mx = fmaxf(mx, t[k]);
        }
        float s = 0.0f;
#pragma unroll
        for (int k = 0; k < KS; ++k) {
            t[k] = __expf(t[k] - mx);
            s += t[k];
        }
        const float inv_s = 1.0f / s;

        // acc[o] += sum_k q_k * wx_nb[k][o]; 9 coalesced 128B L2 loads
#pragma unroll
        for (int k = 0; k < KS; ++k) {
            acc = fmaf(t[k] * inv_s, nrow[k * OUT + lane], acc);
        }
    }

    float res = (deg > 0 ? acc / (float)deg : 0.0f) + bias;
    out[(size_t)w * OUT + lane] = fmaxf(res, 0.0f);
}

// ---------------------------------------------------------------------------
extern "C" void kernel_launch(void* const* d_in, const int* in_sizes, int n_in,
                              void* d_out, int out_size, void* d_ws, size_t ws_size,
                              hipStream_t stream) {
    const float* x    = (const float*)d_in[0];  // (B,V,C)
    const float* W    = (const float*)d_in[1];  // (C,KS,OUT)
    const float* u    = (const float*)d_in[2];  // (C,KS)
    const float* cv   = (const float*)d_in[3];  // (KS,)
    const float* bv   = (const float*)d_in[4];  // (OUT,)
    const int*   adj  = (const int*)d_in[5];    // (V,NB)
    float*       out  = (float*)d_out;          // (B,V,OUT)

    float* Waug = (float*)d_ws;                              // 32*304*4  = ~39 KB
    float* buf  = (float*)((char*)d_ws + 65536);             // 40000*304*4 = ~48.6 MB (L2-resident)

    // Pass 0: augmented weights
    nl_prep_waug<<<(CC * NAUG + 255) / 256, 256, 0, stream>>>(W, u, Waug);

    // Pass 1: WMMA GEMM -> buf (wx | ux)
    const int tiles = MT * NT;                 // 2500 * 19
    nl_gemm_wmma<<<(tiles + 7) / 8, 256, 0, stream>>>(x, Waug, buf);

    // Pass 2: gather + softmax + aggregate
    nl_gather_agg<<<(BB * VV + 7) / 8, 256, 0, stream>>>(buf, adj, cv, bv, out);
}